// ReadingOrderHead_79671643341418
// MI455X (gfx1250) — compile-verified
//
#include <hip/hip_runtime.h>

typedef float v2f __attribute__((ext_vector_type(2)));
typedef float v8f __attribute__((ext_vector_type(8)));

#define DDIM 512
#define NROW 512
#define LDS_STRIDE 516   // 512 + 4 floats pad: row stride shifts bank by 4 -> conflict-free

__global__ __launch_bounds__(256, 2)
void bilinear_score_wmma(const float* __restrict__ X,
                         const float* __restrict__ W,
                         const float* __restrict__ bias_p,
                         float* __restrict__ out) {
  __shared__ float sX[16 * LDS_STRIDE];   // X row panel  (16 x 512)
  __shared__ float sY[16 * LDS_STRIDE];   // Y = Xpanel@W (16 x 512)

  const int tid  = threadIdx.x;
  const int wave = tid >> 5;
  const int lane = tid & 31;
  const int half = lane >> 4;   // 0: lanes 0-15, 1: lanes 16-31
  const int l16  = lane & 15;

  const int wg = blockIdx.x;         // 0..1023
  const int b  = wg >> 5;            // batch index
  const int i0 = (wg & 31) * 16;     // row-tile base within batch

  const float bias = bias_p[0];
  const float* __restrict__ Xb = X + (size_t)b * NROW * DDIM;

  // ---- stage X row panel (16 x 512 f32) into LDS with float4 loads ----
  {
    const float4* __restrict__ src = (const float4*)(Xb + (size_t)i0 * DDIM);
    #pragma unroll
    for (int v = tid; v < 16 * (DDIM / 4); v += 256) {
      const int r = v / (DDIM / 4);
      const int c = v % (DDIM / 4);
      const float4 d = src[r * (DDIM / 4) + c];
      float* dst = &sX[r * LDS_STRIDE + c * 4];
      dst[0] = d.x; dst[1] = d.y; dst[2] = d.z; dst[3] = d.w;
    }
  }
  __syncthreads();

  // ---- phase 1: Y = Xpanel @ W ; wave owns column tiles (wave + 8t), t=0..3 ----
  {
    v8f acc[4];
    #pragma unroll
    for (int t = 0; t < 4; ++t) acc[t] = (v8f)0.f;

    int cbase[4];
    #pragma unroll
    for (int t = 0; t < 4; ++t) cbase[t] = (wave + 8 * t) * 16 + l16;

    const float* __restrict__ sArow = &sX[l16 * LDS_STRIDE];

    #pragma unroll 4
    for (int k0 = 0; k0 < DDIM; k0 += 4) {
      const int kb = k0 + 2 * half;               // K pair handled by this half-wave
      const v2f a = *(const v2f*)(&sArow[kb]);    // ds_load_b64
      #pragma unroll
      for (int t = 0; t < 4; ++t) {
        v2f bb;                                   // B[k,n] = W[k, cbase]
        bb.x = W[(size_t)kb * DDIM + cbase[t]];
        bb.y = W[(size_t)(kb + 1) * DDIM + cbase[t]];
        acc[t] = __builtin_amdgcn_wmma_f32_16x16x4_f32(
            false, a, false, bb, (short)0, acc[t], false, false);
      }
    }

    // scatter D-matrix layout (VGPR r -> row r+8*half, col = lane&15) into sY
    #pragma unroll
    for (int t = 0; t < 4; ++t) {
      const int c = (wave + 8 * t) * 16 + l16;
      #pragma unroll
      for (int r = 0; r < 8; ++r)
        sY[(r + 8 * half) * LDS_STRIDE + c] = acc[t][r];
    }
  }
  __syncthreads();

  // ---- phase 2: S_panel = Y @ Xb^T ; wave owns j-tiles (wave + 8t) ----
  {
    const float* __restrict__ sArow = &sY[l16 * LDS_STRIDE];

    v8f acc[4];
    #pragma unroll
    for (int t = 0; t < 4; ++t) acc[t] = (v8f)0.f;

    int jn[4];
    #pragma unroll
    for (int t = 0; t < 4; ++t) jn[t] = (wave + 8 * t) * 16 + l16;

    #pragma unroll 4
    for (int k0 = 0; k0 < DDIM; k0 += 4) {
      const int kb = k0 + 2 * half;
      const v2f a = *(const v2f*)(&sArow[kb]);
      #pragma unroll
      for (int t = 0; t < 4; ++t) {
        // B[k,n] = Xb[j0+n, k] -> 8-byte global load per lane (L2 resident)
        const v2f bb = *(const v2f*)(Xb + (size_t)jn[t] * DDIM + kb);
        acc[t] = __builtin_amdgcn_wmma_f32_16x16x4_f32(
            false, a, false, bb, (short)0, acc[t], false, false);
      }
    }

    // epilogue: + bias, zero diagonal, coalesced stores
    float* __restrict__ outb = out + (size_t)b * NROW * NROW;
    #pragma unroll
    for (int t = 0; t < 4; ++t) {
      const int j = jn[t];
      #pragma unroll
      for (int r = 0; r < 8; ++r) {
        const int i = i0 + r + 8 * half;
        const float v = acc[t][r] + bias;
        outb[(size_t)i * NROW + j] = (i == j) ? 0.f : v;
      }
    }
  }
}

extern "C" void kernel_launch(void* const* d_in, const int* in_sizes, int n_in,
                              void* d_out, int out_size, void* d_ws, size_t ws_size,
                              hipStream_t stream) {
  (void)in_sizes; (void)n_in; (void)out_size; (void)d_ws; (void)ws_size;
  const float* X    = (const float*)d_in[0];   // (32, 512, 512) f32
  const float* W    = (const float*)d_in[1];   // (512, 512) f32
  const float* bias = (const float*)d_in[2];   // scalar f32
  float* out        = (float*)d_out;           // (32, 512, 512) f32

  const int num_blocks = 32 * (NROW / 16);     // one WG per (batch, 16-row tile)
  bilinear_score_wmma<<<num_blocks, 256, 0, stream>>>(X, W, bias, out);
}